// ShortConv_13649406066734
// MI455X (gfx1250) — compile-verified
//
#include <hip/hip_runtime.h>
#include <hip/hip_bf16.h>

// ShortConv: per-group RMSNorm(C=1024) -> causal depthwise conv K=4 over T -> SiLU
// B=4 T=4096 G=4 C=1024 fp32. Bandwidth-bound (~512MB @ 23.3TB/s ~ 22us).
// gfx1250 paths: global_load_async_to_lds_b128 ring (ASYNCcnt) + exact s_wait_asynccnt.

constexpr int BB   = 4;
constexpr int TDIM = 4096;
constexpr int GG   = 4;
constexpr int CC   = 1024;
constexpr int KK   = 4;
constexpr float EPSV = 1e-5f;

constexpr int TILE = 64;      // time-steps per block
constexpr int NBUF = 4;       // LDS ring depth
constexpr int NTHR = 128;     // 4 waves
constexpr int CPT  = 8;       // channels per thread (2 x float4)

#define WAIT_ASYNC(n) asm volatile("s_wait_asynccnt " #n ::: "memory")

__device__ __forceinline__ void async_b128(const float* g, unsigned lds) {
    asm volatile("global_load_async_to_lds_b128 %0, %1, off"
                 :: "v"(lds), "v"(g) : "memory");
}

__device__ __forceinline__ float4 mul4s(float4 a, float s) {
    return make_float4(a.x * s, a.y * s, a.z * s, a.w * s);
}
__device__ __forceinline__ float4 mul44(float4 a, float4 b) {
    return make_float4(a.x * b.x, a.y * b.y, a.z * b.z, a.w * b.w);
}
__device__ __forceinline__ float4 fma44(float4 a, float4 b, float4 c) {
    return make_float4(fmaf(a.x, b.x, c.x), fmaf(a.y, b.y, c.y),
                       fmaf(a.z, b.z, c.z), fmaf(a.w, b.w, c.w));
}
__device__ __forceinline__ float silu1(float y) {
    return y * __builtin_amdgcn_rcpf(1.0f + __expf(-y));
}
__device__ __forceinline__ float4 silu4(float4 y) {
    return make_float4(silu1(y.x), silu1(y.y), silu1(y.z), silu1(y.w));
}

__global__ __launch_bounds__(NTHR)
void ShortConv_gfx1250_kernel(const float* __restrict__ x,
                              const float* __restrict__ conv_w,
                              const float* __restrict__ norm_w,
                              float* __restrict__ out) {
    __shared__ float srow[NBUF][CC];              // 16 KB raw-row ring
    __shared__ __align__(16) float spart[2][4];   // double-buffered wave partials

    const int tid = threadIdx.x;                  // 0..127
    const int c0  = tid * CPT;
    const int g   = blockIdx.y;
    const int b   = blockIdx.z;
    const int t0  = blockIdx.x * TILE;
    const int rs  = GG * CC;                      // row stride (elements)

    // ---- one-time: norm weights folded into conv taps, transposed to SoA ----
    const float4 nwA = *(const float4*)(norm_w + g * CC + c0);
    const float4 nwB = *(const float4*)(norm_w + g * CC + c0 + 4);
    const float* wp  = conv_w + (size_t)(g * CC + c0) * KK;
    float4 ch0 = mul4s(*(const float4*)(wp +  0), nwA.x);
    float4 ch1 = mul4s(*(const float4*)(wp +  4), nwA.y);
    float4 ch2 = mul4s(*(const float4*)(wp +  8), nwA.z);
    float4 ch3 = mul4s(*(const float4*)(wp + 12), nwA.w);
    float4 ch4 = mul4s(*(const float4*)(wp + 16), nwB.x);
    float4 ch5 = mul4s(*(const float4*)(wp + 20), nwB.y);
    float4 ch6 = mul4s(*(const float4*)(wp + 24), nwB.z);
    float4 ch7 = mul4s(*(const float4*)(wp + 28), nwB.w);
    const float4 wk0A = make_float4(ch0.x, ch1.x, ch2.x, ch3.x);  // tap k=0 (oldest)
    const float4 wk1A = make_float4(ch0.y, ch1.y, ch2.y, ch3.y);
    const float4 wk2A = make_float4(ch0.z, ch1.z, ch2.z, ch3.z);
    const float4 wk3A = make_float4(ch0.w, ch1.w, ch2.w, ch3.w);
    const float4 wk0B = make_float4(ch4.x, ch5.x, ch6.x, ch7.x);
    const float4 wk1B = make_float4(ch4.y, ch5.y, ch6.y, ch7.y);
    const float4 wk2B = make_float4(ch4.z, ch5.z, ch6.z, ch7.z);
    const float4 wk3B = make_float4(ch4.w, ch5.w, ch6.w, ch7.w);

    const float* xbase = x   + ((size_t)b * TDIM * GG + (size_t)g) * CC + c0;
    float*       obase = out + ((size_t)b * TDIM * GG + (size_t)g) * CC + c0;
    const unsigned lds0 = (unsigned)(size_t)(void*)(&srow[0][c0]);

    auto prefetch = [&](const float* rowptr, int buf) {
        const unsigned l = lds0 + (unsigned)buf * (CC * 4u);
        async_b128(rowptr,     l);
        async_b128(rowptr + 4, l + 16u);
    };

    // normalize row sitting in ring slot `buf`: xa/xb = x * rsqrt(mean(x^2)+eps)
    auto normrow = [&](int buf, int par, float4& xa, float4& xb) {
        const float4 va = *(const float4*)(&srow[buf][c0]);
        const float4 vb = *(const float4*)(&srow[buf][c0 + 4]);
        float ps = va.x*va.x + va.y*va.y + va.z*va.z + va.w*va.w
                 + vb.x*vb.x + vb.y*vb.y + vb.z*vb.z + vb.w*vb.w;
        ps += __shfl_xor(ps, 16, 32);
        ps += __shfl_xor(ps,  8, 32);
        ps += __shfl_xor(ps,  4, 32);
        ps += __shfl_xor(ps,  2, 32);
        ps += __shfl_xor(ps,  1, 32);
        if ((tid & 31) == 0) spart[par][tid >> 5] = ps;
        __syncthreads();
        const float4 pq = *(const float4*)(&spart[par][0]);
        const float inv = __builtin_amdgcn_rsqf(
            ((pq.x + pq.y) + (pq.z + pq.w)) * (1.0f / (float)CC) + EPSV);
        xa = mul4s(va, inv);
        xb = mul4s(vb, inv);
    };

    float4 a0{}, a1{}, a2{}, a3{}, b0{}, b1{}, b2{}, b3{};   // causal history (a0 oldest)

    auto push = [&](float4 xa, float4 xb) {
        a0 = a1; b0 = b1; a1 = a2; b1 = b2; a2 = a3; b2 = b3; a3 = xa; b3 = xb;
    };
    auto emit = [&](float* op) {
        float4 yA = fma44(wk0A, a0, fma44(wk1A, a1, fma44(wk2A, a2, mul44(wk3A, a3))));
        float4 yB = fma44(wk0B, b0, fma44(wk1B, b1, fma44(wk2B, b2, mul44(wk3B, b3))));
        *(float4*)op       = silu4(yA);
        *(float4*)(op + 4) = silu4(yB);
    };

    // ---- prologue: rows t0-3..t0 into ring slots 0..3 (skip causal pad) ----
#pragma unroll
    for (int ii = 0; ii < NBUF; ++ii) {
        const int r = t0 - (KK - 1) + ii;
        if (r >= 0) prefetch(xbase + (size_t)r * rs, ii);
    }

    // ---- warmup: rows t0-3..t0-1 -> history a1..a3 (block-uniform branch) ----
    if (t0 > 0) {
        WAIT_ASYNC(0x6); normrow(0, 0, a1, b1); prefetch(xbase + (size_t)(t0 + 1) * rs, 0);
        WAIT_ASYNC(0x6); normrow(1, 1, a2, b2); prefetch(xbase + (size_t)(t0 + 2) * rs, 1);
        WAIT_ASYNC(0x6); normrow(2, 0, a3, b3); prefetch(xbase + (size_t)(t0 + 3) * rs, 2);
    } else {  // t<0 rows are exact zeros after norm; just keep the ring fed
        prefetch(xbase + (size_t)1 * rs, 0);
        prefetch(xbase + (size_t)2 * rs, 1);
        prefetch(xbase + (size_t)3 * rs, 2);
    }

    // ---- steady: rows t0..t0+59 (i = 3..62), static ring slots, renamed history ----
    const float* xp = xbase + (size_t)(t0 + 4) * rs;  // prefetch cursor (r+4)
    float*       op = obase + (size_t)t0 * rs;
#pragma unroll 1
    for (int m = 0; m < (TILE - 4) / 4; ++m) {
#pragma unroll
        for (int ii = 0; ii < 4; ++ii) {
            const int buf = (3 + ii) & 3;
            const int par = (3 + ii) & 1;
            WAIT_ASYNC(0x6);                   // 2 issues/row * 3 rows in flight
            float4 xa, xb;
            normrow(buf, par, xa, xb);
            push(xa, xb);
            prefetch(xp, buf);                 // refill slot just consumed (DS in-order => safe)
            emit(op);
            xp += rs;
            op += rs;
        }
    }

    // ---- tail: rows t0+60..t0+63, exact draining waits, no prefetch ----
    {
        float4 xa, xb;
        WAIT_ASYNC(0x6); normrow(3, 1, xa, xb); push(xa, xb); emit(op); op += rs;
        WAIT_ASYNC(0x4); normrow(0, 0, xa, xb); push(xa, xb); emit(op); op += rs;
        WAIT_ASYNC(0x2); normrow(1, 1, xa, xb); push(xa, xb); emit(op); op += rs;
        WAIT_ASYNC(0x0); normrow(2, 0, xa, xb); push(xa, xb); emit(op);
    }
}

extern "C" void kernel_launch(void* const* d_in, const int* in_sizes, int n_in,
                              void* d_out, int out_size, void* d_ws, size_t ws_size,
                              hipStream_t stream) {
    (void)in_sizes; (void)n_in; (void)out_size; (void)d_ws; (void)ws_size;
    const float* x      = (const float*)d_in[0];  // (B,T,G,C)
    const float* conv_w = (const float*)d_in[1];  // (G*C,1,K)
    const float* norm_w = (const float*)d_in[2];  // (G,C)
    float* out = (float*)d_out;                   // (B,T,G,C)

    dim3 grid(TDIM / TILE, GG, BB);               // 64 x 4 x 4 = 1024 blocks, 4 waves each
    ShortConv_gfx1250_kernel<<<grid, NTHR, 0, stream>>>(x, conv_w, norm_w, out);
}